// GNNModel_71605694759283
// MI455X (gfx1250) — compile-verified
//
#include <hip/hip_runtime.h>

// ---------------------------------------------------------------------------
// 2-layer GCN (PyG GCNConv semantics) for MI455X / gfx1250.
//   out = relu( D^-1/2 (A+I) D^-1/2 (x W) + b )  twice.
// GEMM uses V_WMMA_F32_16X16X4_F32 (full fp32 precision; GEMM is ~1.6 GFLOP
// total and nowhere near the bottleneck). Edge gather/scatter is the memory-
// bound part; h/agg (25.6 MB each) fit in the 192 MB L2, and scatters use
// hardware global_atomic_add_f32 which executes at L2.
// ---------------------------------------------------------------------------

typedef __attribute__((ext_vector_type(2))) float v2f;
typedef __attribute__((ext_vector_type(8))) float v8f;

static __device__ __forceinline__ void atomAddF32(float* p, float v) {
  unsafeAtomicAdd(p, v);  // hardware global_atomic_add_f32 (no CAS loop)
}

// ---- degree / dinv --------------------------------------------------------
__global__ __launch_bounds__(256) void k_zero_f32(float* __restrict__ p, int n) {
  int i = blockIdx.x * 256 + threadIdx.x;
  if (i < n) p[i] = 0.0f;
}

__global__ __launch_bounds__(256) void k_degree(const long long* __restrict__ dst,
                                                float* __restrict__ deg, int nE) {
  int e = blockIdx.x * 256 + threadIdx.x;
  if (e < nE) atomAddF32(&deg[(int)dst[e]], 1.0f);
}

__global__ __launch_bounds__(256) void k_dinv(float* __restrict__ d, int n) {
  int i = blockIdx.x * 256 + threadIdx.x;
  if (i < n) d[i] = rsqrtf(d[i] + 1.0f);  // +1 for self-loop
}

// ---- WMMA GEMM: H[n x 64] = X[n x 64] * W[64 x 64] ------------------------
// One wave computes a 16x64 tile: 4 C-fragments, 16 k-steps of K=4.
// f32 A 16x4 layout (ISA 7.12.2): lanes 0-15 hold K=0(v0),K=1(v1);
//                                 lanes 16-31 hold K=2(v0),K=3(v1).
// f32 C/D 16x16: VGPR j -> row j (lanes 0-15) / row 8+j (lanes 16-31).
__global__ __launch_bounds__(256) void k_gemm_wmma(const float* __restrict__ X,
                                                   const float* __restrict__ W,
                                                   float* __restrict__ H,
                                                   int nrows) {
  const int lane  = threadIdx.x & 31;
  const int wave  = threadIdx.x >> 5;
  const int mtile = blockIdx.x * 8 + wave;
  const int m0    = mtile * 16;
  if (m0 >= nrows) return;          // wave-uniform: EXEC stays all-ones

  const int half = lane >> 4;       // 0: lanes 0-15, 1: lanes 16-31
  const int r    = lane & 15;

  v8f c0 = {}, c1 = {}, c2 = {}, c3 = {};
  const float* xrow = X + (size_t)(m0 + r) * 64;

#pragma unroll
  for (int k0 = 0; k0 < 64; k0 += 4) {
    v2f a;
    a.x = xrow[k0 + 2 * half];
    a.y = xrow[k0 + 2 * half + 1];

    const float* wr0 = W + (size_t)(k0 + 2 * half) * 64;      // row K
    const float* wr1 = wr0 + 64;                              // row K+1
    v2f b0, b1, b2, b3;
    b0.x = wr0[r];       b0.y = wr1[r];
    b1.x = wr0[16 + r];  b1.y = wr1[16 + r];
    b2.x = wr0[32 + r];  b2.y = wr1[32 + r];
    b3.x = wr0[48 + r];  b3.y = wr1[48 + r];

    c0 = __builtin_amdgcn_wmma_f32_16x16x4_f32(false, a, false, b0, (short)0, c0, false, false);
    c1 = __builtin_amdgcn_wmma_f32_16x16x4_f32(false, a, false, b1, (short)0, c1, false, false);
    c2 = __builtin_amdgcn_wmma_f32_16x16x4_f32(false, a, false, b2, (short)0, c2, false, false);
    c3 = __builtin_amdgcn_wmma_f32_16x16x4_f32(false, a, false, b3, (short)0, c3, false, false);
  }

  float* hb = H + (size_t)(m0 + 8 * half) * 64 + r;
#pragma unroll
  for (int j = 0; j < 8; ++j) {
    hb[(size_t)j * 64 +  0] = c0[j];
    hb[(size_t)j * 64 + 16] = c1[j];
    hb[(size_t)j * 64 + 32] = c2[j];
    hb[(size_t)j * 64 + 48] = c3[j];
  }
}

// ---- agg = h * dinv^2  (self-loop seed, no atomics needed) ----------------
__global__ __launch_bounds__(256) void k_init_agg(const float* __restrict__ h,
                                                  const float* __restrict__ dinv,
                                                  float* __restrict__ agg, int nElem) {
  int i = blockIdx.x * 256 + threadIdx.x;
  if (i < nElem) {
    float di = dinv[i >> 6];
    agg[i] = h[i] * di * di;
  }
}

// ---- edge scatter: agg[dst] += h[src] * dinv[src]*dinv[dst] ---------------
// One wave per edge; lane handles 2 contiguous features (256 B/row total).
__global__ __launch_bounds__(256) void k_edge_scatter(const float* __restrict__ H,
                                                      const long long* __restrict__ src,
                                                      const long long* __restrict__ dst,
                                                      const float* __restrict__ dinv,
                                                      float* __restrict__ agg, int nE) {
  const int lane = threadIdx.x & 31;
  const int e    = blockIdx.x * 8 + (threadIdx.x >> 5);
  if (e >= nE) return;
  const int s = (int)src[e];
  const int d = (int)dst[e];
  const float norm = dinv[s] * dinv[d];
  const float2 hv = *(const float2*)(H + (size_t)s * 64 + 2 * lane);
  float* out = agg + (size_t)d * 64 + 2 * lane;
  atomAddF32(out,     hv.x * norm);
  atomAddF32(out + 1, hv.y * norm);
}

// ---- out = relu(agg + b) --------------------------------------------------
__global__ __launch_bounds__(256) void k_finalize(const float* __restrict__ agg,
                                                  const float* __restrict__ bias,
                                                  float* __restrict__ out, int nElem) {
  int i = blockIdx.x * 256 + threadIdx.x;
  if (i < nElem) out[i] = fmaxf(agg[i] + bias[i & 63], 0.0f);
}

// ---------------------------------------------------------------------------
extern "C" void kernel_launch(void* const* d_in, const int* in_sizes, int n_in,
                              void* d_out, int out_size, void* d_ws, size_t ws_size,
                              hipStream_t stream) {
  const float*     x   = (const float*)d_in[0];
  const long long* ei  = (const long long*)d_in[1];   // int64 edge_index [2, E]
  const float*     W1  = (const float*)d_in[2];
  const float*     b1  = (const float*)d_in[3];
  const float*     W2  = (const float*)d_in[4];
  const float*     b2  = (const float*)d_in[5];
  float*           out = (float*)d_out;

  const int Nn = in_sizes[0] / 64;      // 100000 (multiple of 16)
  const int E  = in_sizes[1] / 2;       // 1600000
  const int nElem = Nn * 64;
  const long long* srcp = ei;
  const long long* dstp = ei + E;

  // Workspace carve-out (256 B aligned): dinv | h | agg  (~51.6 MB)
  char* ws = (char*)d_ws;
  auto align256 = [](size_t v) { return (v + 255) & ~(size_t)255; };
  float* dinv = (float*)ws;
  float* h    = (float*)(ws + align256((size_t)Nn * 4));
  float* agg  = (float*)((char*)h + align256((size_t)nElem * 4));
  float* y1   = out;                    // layer-1 activations live in d_out

  const int bElemN  = (Nn + 255) / 256;
  const int bElemE  = (E + 255) / 256;
  const int bElemNF = (nElem + 255) / 256;
  const int bGemm   = ((Nn / 16) + 7) / 8;
  const int bEdge   = (E + 7) / 8;

  // Degree / normalization (shared by both layers)
  k_zero_f32<<<bElemN, 256, 0, stream>>>(dinv, Nn);
  k_degree<<<bElemE, 256, 0, stream>>>(dstp, dinv, E);
  k_dinv<<<bElemN, 256, 0, stream>>>(dinv, Nn);

  // ---- Layer 1 ----
  k_gemm_wmma<<<bGemm, 256, 0, stream>>>(x, W1, h, Nn);
  k_init_agg<<<bElemNF, 256, 0, stream>>>(h, dinv, agg, nElem);
  k_edge_scatter<<<bEdge, 256, 0, stream>>>(h, srcp, dstp, dinv, agg, E);
  k_finalize<<<bElemNF, 256, 0, stream>>>(agg, b1, y1, nElem);

  // ---- Layer 2 ----
  k_gemm_wmma<<<bGemm, 256, 0, stream>>>(y1, W2, h, Nn);
  k_init_agg<<<bElemNF, 256, 0, stream>>>(h, dinv, agg, nElem);
  k_edge_scatter<<<bEdge, 256, 0, stream>>>(h, srcp, dstp, dinv, agg, E);
  k_finalize<<<bElemNF, 256, 0, stream>>>(agg, b2, out, nElem);
}